// RetinaLayer_66194035966259
// MI455X (gfx1250) — compile-verified
//
#include <hip/hip_runtime.h>
#include <hip/hip_bf16.h>

typedef __attribute__((ext_vector_type(2))) float v2f;
typedef __attribute__((ext_vector_type(8))) float v8f;

#define NCLS 80
#define NA   9
#define NH   64
#define NW   64
#define NGT  32
#define NB   8
#define NPER (NA * NH * NW)   /* 36864 anchors per image */
#define NTOT (NB * NPER)      /* 294912 anchors total    */

/* ws layout (float units):
   [0]        loss_xywh accumulator
   [1..8]     num_pos per image
   [9..16]    im_loss_cls per image
   [64 .. 64+NTOT)          penalty weight per anchor (0/1)
   [64+NTOT .. 64+2*NTOT)   (int) matched class if positive else -1          */

__global__ void init_ws(float* ws) {
    if (threadIdx.x < 32) ws[threadIdx.x] = 0.0f;
}

__global__ __launch_bounds__(256) void assign_kernel(
    const float* __restrict__ t_xywh,
    const float* __restrict__ gt_bboxes,
    const int*   __restrict__ gt_cats,
    float* __restrict__ ws)
{
    __shared__ float sGT[NGT * 4];
    __shared__ int   sGC[NGT];
    __shared__ float sLoss;
    __shared__ float sNp;

    const int g = blockIdx.x * 256 + threadIdx.x;  // global anchor id
    const int b = g / NPER;                        // uniform per block (256 | 36864)
    const int n = g % NPER;

    if (threadIdx.x < NGT * 4) sGT[threadIdx.x] = gt_bboxes[b * NGT * 4 + threadIdx.x];
    if (threadIdx.x < NGT)     sGC[threadIdx.x] = gt_cats[b * NGT + threadIdx.x];
    if (threadIdx.x == 0) { sLoss = 0.0f; sNp = 0.0f; }
    __syncthreads();

    const int a  = n / (NH * NW);
    const int hw = n % (NH * NW);
    const int h  = hw / NW;
    const int w  = hw % NW;

    const float sc[3]  = {1.0f, 1.2599f, 1.5874f};
    const float rr0[3] = {1.0f, 1.4f, 0.7f};
    const float rr1[3] = {1.0f, 0.7f, 1.4f};
    const int si = a / 3, ri = a % 3;
    const float aw  = 32.0f * sc[si] * rr0[ri];
    const float ah  = 32.0f * sc[si] * rr1[ri];
    const float acx = ((float)w + 0.5f) * 8.0f;
    const float acy = ((float)h + 0.5f) * 8.0f;

    const float atlx = acx - aw * 0.5f, atly = acy - ah * 0.5f;
    const float abrx = acx + aw * 0.5f, abry = acy + ah * 0.5f;
    const float areaA = aw * ah;

    float best = -1.0f; int bidx = 0;
    for (int k = 0; k < NGT; ++k) {
        const float gcx = sGT[k * 4 + 0], gcy = sGT[k * 4 + 1];
        const float gw  = sGT[k * 4 + 2], gh  = sGT[k * 4 + 3];
        float ix = fminf(abrx, gcx + gw * 0.5f) - fmaxf(atlx, gcx - gw * 0.5f);
        float iy = fminf(abry, gcy + gh * 0.5f) - fmaxf(atly, gcy - gh * 0.5f);
        ix = fmaxf(ix, 0.0f); iy = fmaxf(iy, 0.0f);
        const float inter = ix * iy;
        const float iou   = inter / (areaA + gw * gh - inter);
        if (iou > best) { best = iou; bidx = k; }
    }

    const bool pos = best > 0.5f;
    const bool neg = best < 0.4f;
    const bool pen = pos || neg;

    if (pos) {
        const float gcx = sGT[bidx * 4 + 0], gcy = sGT[bidx * 4 + 1];
        const float gw  = sGT[bidx * 4 + 2], gh  = sGT[bidx * 4 + 3];
        const float tx = (gcx - acx) / aw;
        const float ty = (gcy - acy) / ah;
        const float tw = __logf(gw / aw + 1e-8f);   // v_log_f32 hardware trans
        const float th = __logf(gh / ah + 1e-8f);
        const float4 t4 = reinterpret_cast<const float4*>(t_xywh)[g];
        const float d0 = t4.x - tx, d1 = t4.y - ty, d2 = t4.z - tw, d3 = t4.w - th;
        atomicAdd(&sLoss, d0 * d0 + d1 * d1 + d2 * d2 + d3 * d3);
        atomicAdd(&sNp, 1.0f);
    }

    float* penf  = ws + 64;
    int*   posgc = reinterpret_cast<int*>(ws + 64 + NTOT);
    penf[g]  = pen ? 1.0f : 0.0f;
    posgc[g] = pos ? sGC[bidx] : -1;

    __syncthreads();
    if (threadIdx.x == 0) {
        atomicAdd(&ws[0], sLoss);
        atomicAdd(&ws[1 + b], sNp);
    }
}

/* Penalty-weighted softplus: pen * (max(x,0) + log(1 + e^{-|x|})) using the
   hardware v_exp_f32 / v_log_f32 trans pipe (exp(-|x|) in (0,1] => safe). */
__device__ __forceinline__ float bce_term(float x, float pen, bool hit) {
    float v = fmaxf(x, 0.0f) + __logf(1.0f + __expf(-fabsf(x)));
    if (hit) v -= x;            // -x[gc] at positive anchors
    return pen * v;
}

/* Each wave: 16-anchor tile x 80 classes.  Class-dim reduction on the matrix
   pipe with V_WMMA_F32_16X16X4_F32, B = ones.  Since B is all-ones the K
   assignment inside the 16x4 A tile is order-free, so each lane loads a full
   float4 (lane-half 0: classes 8i..8i+3, half 1: 8i+4..8i+7) and feeds two
   WMMAs per iteration -> b128 loads on the 94MB roofline-critical stream. */
__global__ __launch_bounds__(256) void cls_loss_kernel(
    const float* __restrict__ cls_logits,
    float* __restrict__ ws)
{
    __shared__ float sSum;
    if (threadIdx.x == 0) sSum = 0.0f;
    __syncthreads();

    const float* penf  = ws + 64;
    const int*   posgc = reinterpret_cast<const int*>(ws + 64 + NTOT);

    const int lane  = threadIdx.x & 31;
    const int wave  = threadIdx.x >> 5;
    const int m     = lane & 15;        // anchor row within tile
    const int khalf = lane >> 4;        // which half of each 8-class chunk
    const int g     = blockIdx.x * 128 + wave * 16 + m;

    const float pen = penf[g];
    const int   pg  = posgc[g];

    const float* base = cls_logits + (size_t)g * NCLS;

    v8f acc = {0.f, 0.f, 0.f, 0.f, 0.f, 0.f, 0.f, 0.f};
    const v2f bones = {1.0f, 1.0f};

    for (int i = 0; i < NCLS / 8; ++i) {
        const int c = 8 * i + 4 * khalf;
        __builtin_prefetch(base + c + 64, 0, 1);   // global_prefetch_b8 ahead of stream
        const float4 x4 = *reinterpret_cast<const float4*>(base + c);
        const v2f a0 = { bce_term(x4.x, pen, pg == c),
                         bce_term(x4.y, pen, pg == c + 1) };
        const v2f a1 = { bce_term(x4.z, pen, pg == c + 2),
                         bce_term(x4.w, pen, pg == c + 3) };
        acc = __builtin_amdgcn_wmma_f32_16x16x4_f32(false, a0, false, bones,
                                                    (short)0, acc, false, false);
        acc = __builtin_amdgcn_wmma_f32_16x16x4_f32(false, a1, false, bones,
                                                    (short)0, acc, false, false);
    }

    /* B==ones => all 16 D columns identical: lanes 0-15 hold rows 0-7 sums,
       lanes 16-31 hold rows 8-15.  Tile total = t(lane0) + t(lane16). */
    const float t = acc[0] + acc[1] + acc[2] + acc[3]
                  + acc[4] + acc[5] + acc[6] + acc[7];
    if ((lane & 15) == 0) atomicAdd(&sSum, t);

    __syncthreads();
    if (threadIdx.x == 0) {
        const int b = blockIdx.x / (NPER / 128);  // 288 blocks per image
        atomicAdd(&ws[9 + b], sSum);
    }
}

__global__ void finalize_kernel(const float* __restrict__ ws, float* __restrict__ out) {
    if (threadIdx.x == 0) {
        float s = ws[0];  // loss_xywh
        for (int b = 0; b < NB; ++b) s += ws[9 + b] / (ws[1 + b] + 1.0f);
        out[0] = s / (float)NB;
    }
}

extern "C" void kernel_launch(void* const* d_in, const int* in_sizes, int n_in,
                              void* d_out, int out_size, void* d_ws, size_t ws_size,
                              hipStream_t stream) {
    (void)in_sizes; (void)n_in; (void)out_size; (void)ws_size;
    const float* t_xywh     = (const float*)d_in[0];
    const float* cls_logits = (const float*)d_in[1];
    const float* gt_bboxes  = (const float*)d_in[2];
    const int*   gt_cats    = (const int*)d_in[3];
    float* ws  = (float*)d_ws;
    float* out = (float*)d_out;

    init_ws<<<1, 32, 0, stream>>>(ws);
    assign_kernel<<<NTOT / 256, 256, 0, stream>>>(t_xywh, gt_bboxes, gt_cats, ws);
    cls_loss_kernel<<<NTOT / 128, 256, 0, stream>>>(cls_logits, ws);
    finalize_kernel<<<1, 32, 0, stream>>>(ws, out);
}